// KGGraphAttentionLayer_10943576670967
// MI455X (gfx1250) — compile-verified
//
#include <hip/hip_runtime.h>
#include <hip/hip_bf16.h>

typedef __attribute__((ext_vector_type(2))) float v2f;
typedef __attribute__((ext_vector_type(8))) float v8f;

#define ALPHA 0.2f
#define D 128

// -------- helper: fast f32 global atomic add (lowers to global_atomic_add_f32) ----
__device__ __forceinline__ void atomAddF32(float* p, float v) {
#if defined(__AMDGCN__)
    unsafeAtomicAdd(p, v);
#else
    atomicAdd(p, v);
#endif
}

// ============================================================================
// Kernel 1: row scores via V_WMMA_F32_16X16X4_F32.
// One wave32 per 16-row tile. Stage tile into LDS coalesced, then 32 chained
// WMMAs over K=128. B-matrix = attn chunk replicated across all 16 columns,
// so every output column equals the dot product; we read column N=0
// (lane 0 -> rows M=0..7 in c[0..7], lane 16 -> rows M=8..15).
// ============================================================================
__global__ void __launch_bounds__(32)
row_scores_wmma(const float* __restrict__ head_rep,
                const float* __restrict__ tail_rep,
                const float* __restrict__ attn,
                float* __restrict__ ws,      // ws[0..B)=ah, ws[B..B+N)=at
                int B, int N, int headTiles) {
    __shared__ float tile[16 * D];           // 8 KB
    const int lane = threadIdx.x;            // 0..31 (wave32)
    const int t    = blockIdx.x;

    const float* rep;  const float* aseg;  float* outp;  int rows, row0;
    if (t < headTiles) { rep = head_rep; aseg = attn;       outp = ws;     rows = B; row0 = t * 16; }
    else               { rep = tail_rep; aseg = attn + D;   outp = ws + B; rows = N; row0 = (t - headTiles) * 16; }

    // ---- stage 16 rows x 128 floats into LDS, fully coalesced float4 ----
    #pragma unroll 4
    for (int i = 0; i < 16; ++i) {
        int r = row0 + i; if (r >= rows) r = rows - 1;     // clamp (padding rows discarded at store)
        float4 v = ((const float4*)(rep + (size_t)r * D))[lane];
        ((float4*)tile)[i * 32 + lane] = v;
    }
    __syncthreads();   // single-wave WG: barrier -> s_nop, keeps the dscnt wait

    // ---- 32 x V_WMMA_F32_16X16X4_F32, accumulating over K ----
    const int half = lane >> 4;        // A layout: low lanes K=0,1 ; high lanes K=2,3
    const int m    = lane & 15;        // A row M
    v8f c = {};
    #pragma unroll
    for (int kk = 0; kk < 32; ++kk) {
        const int kb = kk * 4 + half * 2;
        v2f a, b;
        a.x = tile[m * D + kb];
        a.y = tile[m * D + kb + 1];
        b.x = aseg[kb];                // replicate attn value across all N columns
        b.y = aseg[kb + 1];
        c = __builtin_amdgcn_wmma_f32_16x16x4_f32(
                /*neg_a=*/false, a, /*neg_b=*/false, b,
                /*c_mod=*/(short)0, c, /*reuse_a=*/false, /*reuse_b=*/false);
    }

    // ---- column N=0 of D: lane 0 holds M=0..7, lane 16 holds M=8..15 ----
    if (m == 0) {
        const int base = row0 + half * 8;
        #pragma unroll
        for (int i = 0; i < 8; ++i) {
            const int r = base + i;
            if (r < rows) outp[r] = c[i];
        }
    }
}

// ============================================================================
// Kernel 2: zero the output (e_rowsum [B] ++ h_prime [B*D]); required each
// call since kernel 3 accumulates with atomics.
// ============================================================================
__global__ void zero_out(float* __restrict__ p, int n) {
    const int i = blockIdx.x * blockDim.x + threadIdx.x;
    if (i < n) p[i] = 0.0f;
}

// ============================================================================
// Kernel 3: one wave32 per edge.
//   w = exp(leakyrelu(clip(ah[eh]+at[et], -10, 10)))
//   lane l: float4 of tail_val row et -> 4 atomic f32 adds into h_prime row eh
//   lane 0: atomic add w into e_rowsum[eh]
// Row gathers are one coalesced 512B request per wave; tail_val / h_prime are
// L2-resident (51 MB each vs 192 MB L2).
// ============================================================================
__global__ void __launch_bounds__(256)
edge_scatter(const int* __restrict__ edges,    // [2, E] flat: heads, then tails
             const float* __restrict__ tail_val,
             const float* __restrict__ scores, // ah[0..B), at[B..B+N)
             float* __restrict__ out,          // e_rowsum[0..B), h_prime[B..B+B*D)
             int B, int E) {
    const int tid  = blockIdx.x * 256 + threadIdx.x;
    const int e    = tid >> 5;
    const int lane = tid & 31;
    if (e >= E) return;

    const int eh = edges[e];
    const int et = edges[E + e];

    float s = scores[eh] + scores[B + et];
    s = fminf(10.0f, fmaxf(-10.0f, s));
    const float l = (s > 0.0f) ? s : ALPHA * s;
    const float w = __expf(l);

    const float4 v = ((const float4*)(tail_val + (size_t)et * D))[lane];
    float* hp = out + B + (size_t)eh * D + lane * 4;
    atomAddF32(hp + 0, w * v.x);
    atomAddF32(hp + 1, w * v.y);
    atomAddF32(hp + 2, w * v.z);
    atomAddF32(hp + 3, w * v.w);

    if (lane == 0) atomAddF32(out + eh, w);
}

// ============================================================================
extern "C" void kernel_launch(void* const* d_in, const int* in_sizes, int n_in,
                              void* d_out, int out_size, void* d_ws, size_t ws_size,
                              hipStream_t stream) {
    const float* head_rep = (const float*)d_in[0];
    const float* tail_rep = (const float*)d_in[1];
    const float* tail_val = (const float*)d_in[2];
    const int*   edges    = (const int*)d_in[3];
    const float* attn     = (const float*)d_in[4];

    const int B = in_sizes[0] / D;
    const int N = in_sizes[1] / D;
    const int E = in_sizes[3] / 2;

    float* ws  = (float*)d_ws;    // (B + N) floats of score workspace
    float* out = (float*)d_out;   // B + B*D floats

    // 1) row scores (WMMA GEMV): one wave32 block per 16-row tile
    const int headTiles = (B + 15) / 16;
    const int tailTiles = (N + 15) / 16;
    row_scores_wmma<<<headTiles + tailTiles, 32, 0, stream>>>(
        head_rep, tail_rep, attn, ws, B, N, headTiles);

    // 2) zero outputs (accumulated via atomics)
    const int total = B + B * D;
    zero_out<<<(total + 255) / 256, 256, 0, stream>>>(out, total);

    // 3) edge scatter: one wave per edge, 8 waves per 256-thread block
    const long long threads = (long long)E * 32;
    const int blocks = (int)((threads + 255) / 256);
    edge_scatter<<<blocks, 256, 0, stream>>>(edges, tail_val, ws, out, B, E);
}